// GravityDisplacement_60361470378597
// MI455X (gfx1250) — compile-verified
//
#include <hip/hip_runtime.h>
#include <math.h>

typedef __attribute__((ext_vector_type(2))) float v2f;
typedef __attribute__((ext_vector_type(8))) float v8f;

constexpr int Bn   = 2;
constexpr int L    = 2304;
constexpr int D    = 256;
constexpr int HID  = 256;
constexpr int HID2 = 128;
constexpr int ROWS = Bn * L;              // 4608 latent rows total

constexpr float SMINV   = -51.5f;
constexpr float SMAXV   =  51.5f;
constexpr float SPACING = 103.0f / 47.0f; // surface/(row-1)
constexpr float DANGER  = SPACING * 0.5f;
constexpr float SIGMA   = SPACING * 0.5f;
constexpr float STEPC   = SPACING * 0.1f;
constexpr float MAX_STEP  = SPACING * 0.25f;
constexpr float MAX_TOTAL = SPACING * 0.5f;
constexpr float EPSV    = 1e-8f;

__device__ __forceinline__ float gelu_exact(float x) {
  return 0.5f * x * (1.0f + erff(x * 0.70710678118654752440f));
}
__device__ __forceinline__ float softplus_f(float x) {
  return (x > 20.0f) ? x : log1pf(expf(x));
}

// ---------------------------------------------------------------------------
// Fused MLP error predictor:  errors = softplus(g2(g1(LN(x@W1+b1))@W2+b2)@W3+b3)
// One block = 16 rows; 8 waves; fp32 WMMA 16x16x4 for both GEMMs.
// ---------------------------------------------------------------------------
__global__ __launch_bounds__(256) void mlp_errors_kernel(
    const float* __restrict__ latents, const float* __restrict__ W1,
    const float* __restrict__ b1, const float* __restrict__ gamma,
    const float* __restrict__ beta, const float* __restrict__ W2,
    const float* __restrict__ b2, const float* __restrict__ W3,
    const float* __restrict__ b3, float* __restrict__ errors) {
  __shared__ float lat[16 * 256];
  __shared__ float h1[16 * 256];
  __shared__ float h2[16 * 128];
  __shared__ float mu_s[16], rstd_s[16];

  const int tid  = threadIdx.x;
  const int wave = tid >> 5;
  const int lane = tid & 31;
  const int row0 = blockIdx.x * 16;

  // Stage 16x256 latents tile into LDS (coalesced).
  for (int i = 0; i < 16; ++i)
    lat[tid + i * 256] = latents[(size_t)row0 * D + tid + i * 256];
  __syncthreads();

  // fp32 WMMA lane mapping (ISA 7.12.2): lanes 0-15 hold K0/K1, lanes 16-31 K2/K3.
  const int r    = lane & 15;          // M row (A) / N col (B,C,D) within tile
  const int koff = (lane >> 4) << 1;   // 0 or 2
  const int hi8  = (lane >> 4) << 3;   // C/D: lanes>=16 hold M+8

  // ---- Layer 1: [16x256] @ [256x256], wave w owns cols [32w, 32w+32) ----
  {
    const int n0 = wave * 32;
    v8f c0 = {}; v8f c1 = {};
    for (int k = 0; k < D; k += 4) {
      v2f a;
      a.x = lat[r * 256 + k + koff];
      a.y = lat[r * 256 + k + koff + 1];
      const float* w1p = W1 + (size_t)(k + koff) * HID;
      v2f bA, bB;
      bA.x = w1p[n0 + r];        bA.y = w1p[HID + n0 + r];
      bB.x = w1p[n0 + 16 + r];   bB.y = w1p[HID + n0 + 16 + r];
      c0 = __builtin_amdgcn_wmma_f32_16x16x4_f32(false, a, false, bA, (short)0, c0, false, false);
      c1 = __builtin_amdgcn_wmma_f32_16x16x4_f32(false, a, false, bB, (short)0, c1, false, false);
    }
    for (int j = 0; j < 8; ++j) {
      const int m = j + hi8;
      h1[m * 256 + n0 + r]      = c0[j] + b1[n0 + r];
      h1[m * 256 + n0 + 16 + r] = c1[j] + b1[n0 + 16 + r];
    }
  }
  __syncthreads();

  // ---- LayerNorm statistics per row ----
  if (tid < 16) {
    float s = 0.0f, s2 = 0.0f;
    for (int c = 0; c < 256; ++c) { float v = h1[tid * 256 + c]; s += v; s2 += v * v; }
    const float mu  = s * (1.0f / 256.0f);
    const float var = fmaxf(s2 * (1.0f / 256.0f) - mu * mu, 0.0f);
    mu_s[tid]   = mu;
    rstd_s[tid] = rsqrtf(var + 1e-5f);
  }
  __syncthreads();

  // ---- Apply LN * gamma + beta, then GELU (thread t owns column t) ----
  for (int m = 0; m < 16; ++m) {
    float v = h1[m * 256 + tid];
    v = (v - mu_s[m]) * rstd_s[m] * gamma[tid] + beta[tid];
    h1[m * 256 + tid] = gelu_exact(v);
  }
  __syncthreads();

  // ---- Layer 2: [16x256] @ [256x128], wave w owns cols [16w, 16w+16) ----
  {
    const int n0 = wave * 16;
    v8f c = {};
    for (int k = 0; k < HID; k += 4) {
      v2f a;
      a.x = h1[r * 256 + k + koff];
      a.y = h1[r * 256 + k + koff + 1];
      const float* w2p = W2 + (size_t)(k + koff) * HID2;
      v2f bb;
      bb.x = w2p[n0 + r];
      bb.y = w2p[HID2 + n0 + r];
      c = __builtin_amdgcn_wmma_f32_16x16x4_f32(false, a, false, bb, (short)0, c, false, false);
    }
    for (int j = 0; j < 8; ++j) {
      const int m = j + hi8;
      h2[m * 128 + n0 + r] = gelu_exact(c[j] + b2[n0 + r]);
    }
  }
  __syncthreads();

  // ---- Layer 3: [16x128] @ [128x1] + softplus ----
  if (tid < 16) {
    float acc = 0.0f;
    for (int k = 0; k < HID2; ++k) acc += h2[tid * 128 + k] * W3[k];
    acc += b3[0];
    errors[row0 + tid] = softplus_f(acc);
  }
}

// ---------------------------------------------------------------------------
// Per-batch min/max of errors (for norm01).  mm[b]=min, mm[2+b]=max.
// ---------------------------------------------------------------------------
__global__ __launch_bounds__(256) void minmax_kernel(const float* __restrict__ errors,
                                                     float* __restrict__ mm) {
  __shared__ float smn[256], smx[256];
  const int b = blockIdx.x;
  const int t = threadIdx.x;
  const float* e = errors + b * L;
  float mn = 3.4e38f, mx = -3.4e38f;
  for (int i = t; i < L; i += 256) { float v = e[i]; mn = fminf(mn, v); mx = fmaxf(mx, v); }
  smn[t] = mn; smx[t] = mx;
  __syncthreads();
  for (int s = 128; s > 0; s >>= 1) {
    if (t < s) { smn[t] = fminf(smn[t], smn[t + s]); smx[t] = fmaxf(smx[t], smx[t + s]); }
    __syncthreads();
  }
  if (t == 0) { mm[b] = smn[0]; mm[2 + b] = smx[0]; }
}

// ---------------------------------------------------------------------------
// Phase 1: gravity toward high-error latents + danger-zone repulsion.
// One thread per particle i; j-particles streamed through LDS tiles.
// Also emits strength = 1 - norm01(errors) for phase 2.
// ---------------------------------------------------------------------------
__global__ __launch_bounds__(256) void phase1_kernel(
    const float* __restrict__ positions, const float* __restrict__ errors,
    const float* __restrict__ mm, float* __restrict__ pos1,
    float* __restrict__ strength) {
  __shared__ float pjx[256], pjy[256], ejn[256];
  constexpr int BPB = L / 256;  // 9 blocks per batch
  const int b  = blockIdx.x / BPB;
  const int i  = (blockIdx.x % BPB) * 256 + threadIdx.x;
  const int gi = b * L + i;
  const float emin = mm[b], emax = mm[2 + b];
  const float inv_range = 1.0f / (emax - emin + 1e-8f);
  const float px = positions[2 * gi], py = positions[2 * gi + 1];

  float gx = 0.0f, gy = 0.0f, rx = 0.0f, ry = 0.0f;
  for (int tile = 0; tile < BPB; ++tile) {
    const int gj = b * L + tile * 256 + threadIdx.x;
    pjx[threadIdx.x] = positions[2 * gj];
    pjy[threadIdx.x] = positions[2 * gj + 1];
    ejn[threadIdx.x] = (errors[gj] - emin) * inv_range + 0.1f;
    __syncthreads();
    for (int jj = 0; jj < 256; ++jj) {
      const int jg = tile * 256 + jj;
      if (jg != i) {
        const float dx = pjx[jj] - px, dy = pjy[jj] - py;
        const float dist = fmaxf(sqrtf(dx * dx + dy * dy), 1e-6f);
        const float inv  = 1.0f / dist;
        const float dirx = dx * inv, diry = dy * inv;
        const float g = ejn[jj] * inv * inv;                 // GRAV_P = 2
        gx += dirx * g; gy += diry * g;
        const float viol = fmaxf(DANGER - dist, 0.0f);
        const float rep  = expf(viol * (1.0f / DANGER)) - 1.0f;
        rx -= dirx * rep; ry -= diry * rep;
      }
    }
    __syncthreads();
  }
  const float fx = gx + 0.5f * rx, fy = gy + 0.5f * ry;
  const float fmag  = sqrtf(fx * fx + fy * fy);
  const float scale = fminf(fmaxf(fmag, 0.5f), 3.0f) / (fmag + EPSV);
  pos1[2 * gi]     = fminf(fmaxf(px + fx * scale, SMINV), SMAXV);
  pos1[2 * gi + 1] = fminf(fmaxf(py + fy * scale, SMINV), SMAXV);
  strength[gi]     = 1.0f - (errors[gi] - emin) * inv_range;
}

// ---------------------------------------------------------------------------
// Phase 2 single density-spreading iteration (launched 3x with ping-pong).
// ---------------------------------------------------------------------------
__global__ __launch_bounds__(256) void phase2_kernel(
    const float* __restrict__ posCur, const float* __restrict__ pos0,
    const float* __restrict__ strength, float* __restrict__ posNext) {
  __shared__ float pjx[256], pjy[256];
  constexpr int BPB = L / 256;
  const int b  = blockIdx.x / BPB;
  const int i  = (blockIdx.x % BPB) * 256 + threadIdx.x;
  const int gi = b * L + i;
  const float px = posCur[2 * gi], py = posCur[2 * gi + 1];
  constexpr float inv2s2 = 1.0f / (2.0f * SIGMA * SIGMA);
  constexpr float invs2  = 1.0f / (SIGMA * SIGMA);

  float gradx = 0.0f, grady = 0.0f;
  for (int tile = 0; tile < BPB; ++tile) {
    const int gj = b * L + tile * 256 + threadIdx.x;
    pjx[threadIdx.x] = posCur[2 * gj];
    pjy[threadIdx.x] = posCur[2 * gj + 1];
    __syncthreads();
    for (int jj = 0; jj < 256; ++jj) {
      const int jg = tile * 256 + jj;
      if (jg != i) {
        const float dx = pjx[jj] - px, dy = pjy[jj] - py;
        const float w = expf(-(dx * dx + dy * dy) * inv2s2);
        gradx += w * dx; grady += w * dy;
      }
    }
    __syncthreads();
  }
  gradx *= invs2; grady *= invs2;
  const float gnorm = sqrtf(gradx * gradx + grady * grady);
  const float s = -(STEPC * strength[gi]) / (gnorm + EPSV);
  float sx = gradx * s, sy = grady * s;
  const float snorm = sqrtf(sx * sx + sy * sy);
  const float cap1 = fminf(1.0f, MAX_STEP / (snorm + EPSV));
  sx *= cap1; sy *= cap1;
  const float p0x = pos0[2 * gi], p0y = pos0[2 * gi + 1];
  float tx = (px + sx) - p0x;
  float ty = (py + sy) - p0y;
  const float tnorm = sqrtf(tx * tx + ty * ty);
  const float cap2 = fminf(1.0f, MAX_TOTAL / (tnorm + EPSV));
  tx *= cap2; ty *= cap2;
  posNext[2 * gi]     = fminf(fmaxf(p0x + tx, SMINV), SMAXV);
  posNext[2 * gi + 1] = fminf(fmaxf(p0y + ty, SMINV), SMAXV);
}

// ---------------------------------------------------------------------------
extern "C" void kernel_launch(void* const* d_in, const int* in_sizes, int n_in,
                              void* d_out, int out_size, void* d_ws, size_t ws_size,
                              hipStream_t stream) {
  const float* latents   = (const float*)d_in[0];
  const float* positions = (const float*)d_in[1];
  const float* W1    = (const float*)d_in[2];
  const float* b1    = (const float*)d_in[3];
  const float* gamma = (const float*)d_in[4];
  const float* beta  = (const float*)d_in[5];
  const float* W2    = (const float*)d_in[6];
  const float* b2    = (const float*)d_in[7];
  const float* W3    = (const float*)d_in[8];
  const float* b3    = (const float*)d_in[9];
  float* out = (float*)d_out;

  float* w        = (float*)d_ws;
  float* errors   = w;             // 4608 floats
  float* mm       = w + 4608;      // 4 floats (min[2], max[2])
  float* strength = w + 4616;      // 4608 floats
  float* pos0     = w + 9224;      // 9216 floats (phase-1 result / spread origin)
  float* posB     = w + 18440;     // 9216 floats
  float* posC     = w + 27656;     // 9216 floats

  const int gridP = Bn * (L / 256);  // 18 blocks

  mlp_errors_kernel<<<ROWS / 16, 256, 0, stream>>>(latents, W1, b1, gamma, beta,
                                                   W2, b2, W3, b3, errors);
  minmax_kernel<<<Bn, 256, 0, stream>>>(errors, mm);
  phase1_kernel<<<gridP, 256, 0, stream>>>(positions, errors, mm, pos0, strength);
  phase2_kernel<<<gridP, 256, 0, stream>>>(pos0, pos0, strength, posB);
  phase2_kernel<<<gridP, 256, 0, stream>>>(posB, pos0, strength, posC);
  phase2_kernel<<<gridP, 256, 0, stream>>>(posC, pos0, strength, out);
}